// EdgeGATModel_37598143709435
// MI455X (gfx1250) — compile-verified
//
#include <hip/hip_runtime.h>

typedef __attribute__((ext_vector_type(2))) float v2f;
typedef __attribute__((ext_vector_type(8))) float v8f;

// ---------------------------------------------------------------------------
// float atomic max via monotonic int mapping (init memory to 0xFF800000 = -inf)
// ---------------------------------------------------------------------------
__device__ __forceinline__ void atomicMaxF(float* addr, float v) {
  if (v >= 0.f) atomicMax((int*)addr, __float_as_int(v));
  else          atomicMin((unsigned int*)addr, (unsigned int)__float_as_int(v));
}

// ---------------------------------------------------------------------------
// init: zero accumulators, set segment-max slots to -inf bits
// ---------------------------------------------------------------------------
__global__ void k_init(float* agg1, float* den1, int* amax1,
                       float* agg2, float* den2, int* amax2,
                       float* psum, float* pcnt, int Nn, int Gg) {
  const int NEGINF = (int)0xFF800000u;
  long total = (long)Nn * 256;
  for (long i = blockIdx.x * (long)blockDim.x + threadIdx.x; i < total;
       i += (long)gridDim.x * blockDim.x) {
    agg1[i] = 0.f;
    if (i < (long)Nn * 64) agg2[i] = 0.f;
    if (i < (long)Nn * 4) { den1[i] = 0.f; amax1[i] = NEGINF; }
    if (i < Nn)           { den2[i] = 0.f; amax2[i] = NEGINF; }
    if (i < (long)Gg * 64) psum[i] = 0.f;
    if (i < Gg)            pcnt[i] = 0.f;
  }
}

// ---------------------------------------------------------------------------
// conv1 node projections: x[N,2] -> q1,k1,v1,xr1 [N,256] (K=2: pure VALU)
// ---------------------------------------------------------------------------
__global__ void k_nodeproj1(const float* __restrict__ x,
                            const float* __restrict__ Wq, const float* __restrict__ bq,
                            const float* __restrict__ Wk, const float* __restrict__ bk,
                            const float* __restrict__ Wv, const float* __restrict__ bv,
                            const float* __restrict__ Ws, const float* __restrict__ bs,
                            float* q, float* k, float* v, float* xr, int Nn) {
  long i = blockIdx.x * (long)blockDim.x + threadIdx.x;
  if (i >= (long)Nn * 256) return;
  int n = (int)(i >> 8), c = (int)(i & 255);
  float x0 = x[2 * n], x1 = x[2 * n + 1];
  q[i]  = x0 * Wq[c] + x1 * Wq[256 + c] + bq[c];
  k[i]  = x0 * Wk[c] + x1 * Wk[256 + c] + bk[c];
  v[i]  = x0 * Wv[c] + x1 * Wv[256 + c] + bv[c];
  xr[i] = x0 * Ws[c] + x1 * Ws[256 + c] + bs[c];
}

// ---------------------------------------------------------------------------
// conv1 edge pass A: alpha[e,h] = <q1[dst,h], k1[src,h] + e1[e,h]> / 8
// wave-per-edge (8 edges/block); We1 (18x256, 18KB) staged in LDS;
// e1 recomputed on the fly (cheaper than 410MB materialization).
// ---------------------------------------------------------------------------
__global__ void k_alpha1(const float* __restrict__ q1, const float* __restrict__ k1,
                         const float* __restrict__ ea, const int* __restrict__ srcA,
                         const int* __restrict__ dstA, const float* __restrict__ We,
                         float* alpha, int* amaxB, int Ee) {
  __shared__ float WeS[18 * 256];
  int t = threadIdx.x;
  for (int i = t; i < 18 * 256; i += 256) WeS[i] = We[i];
  __syncthreads();
  int wave = t >> 5, lane = t & 31;
  int eid = blockIdx.x * 8 + wave;
  if (eid >= Ee) return;
  int src = srcA[eid], dst = dstA[eid];
  float av = (lane < 18) ? ea[(long)eid * 18 + lane] : 0.f;
  int c0 = lane * 8;
  float e[8];
#pragma unroll
  for (int cc = 0; cc < 8; cc++) e[cc] = 0.f;
#pragma unroll
  for (int j = 0; j < 18; j++) {
    float a = __shfl(av, j, 32);
    const float* wr = &WeS[j * 256 + c0];
#pragma unroll
    for (int cc = 0; cc < 8; cc++) e[cc] += a * wr[cc];
  }
  const float* kr = k1 + (long)src * 256 + c0;
  const float* qr = q1 + (long)dst * 256 + c0;
  float s = 0.f;
#pragma unroll
  for (int cc = 0; cc < 8; cc++) s += qr[cc] * (kr[cc] + e[cc]);
  s += __shfl_xor(s, 1, 32);
  s += __shfl_xor(s, 2, 32);
  s += __shfl_xor(s, 4, 32);
  if ((lane & 7) == 0) {
    int h = lane >> 3;
    float a = s * 0.125f;
    alpha[(long)eid * 4 + h] = a;
    atomicMaxF((float*)&amaxB[(long)dst * 4 + h], a);
  }
}

// conv1 edge pass B: ex = exp(alpha - amax[dst]); den[dst] += ex
__global__ void k_expden1(float* alpha, const int* __restrict__ dstA,
                          const int* __restrict__ amaxB, float* den, int Ee) {
  long i = blockIdx.x * (long)blockDim.x + threadIdx.x;
  if (i >= (long)Ee * 4) return;
  int eid = (int)(i >> 2), h = (int)(i & 3);
  int dst = dstA[eid];
  float m = __int_as_float(amaxB[(long)dst * 4 + h]);
  float ex = __expf(alpha[i] - m);
  alpha[i] = ex;
  atomicAdd(&den[(long)dst * 4 + h], ex);
}

// conv1 edge pass C: agg[dst] += (v1[src] + e1) * ex/(den[dst]+1e-16)
__global__ void k_msg1(const float* __restrict__ v1, const float* __restrict__ ea,
                       const int* __restrict__ srcA, const int* __restrict__ dstA,
                       const float* __restrict__ We, const float* __restrict__ alpha,
                       const float* __restrict__ den, float* agg, int Ee) {
  __shared__ float WeS[18 * 256];
  int t = threadIdx.x;
  for (int i = t; i < 18 * 256; i += 256) WeS[i] = We[i];
  __syncthreads();
  int wave = t >> 5, lane = t & 31;
  int eid = blockIdx.x * 8 + wave;
  if (eid >= Ee) return;
  int src = srcA[eid], dst = dstA[eid];
  float av = (lane < 18) ? ea[(long)eid * 18 + lane] : 0.f;
  int c0 = lane * 8;
  float e[8];
#pragma unroll
  for (int cc = 0; cc < 8; cc++) e[cc] = 0.f;
#pragma unroll
  for (int j = 0; j < 18; j++) {
    float a = __shfl(av, j, 32);
    const float* wr = &WeS[j * 256 + c0];
#pragma unroll
    for (int cc = 0; cc < 8; cc++) e[cc] += a * wr[cc];
  }
  int h = lane >> 3;
  float w = alpha[(long)eid * 4 + h] / (den[(long)dst * 4 + h] + 1e-16f);
  const float* vr = v1 + (long)src * 256 + c0;
  float* ag = agg + (long)dst * 256 + c0;
#pragma unroll
  for (int cc = 0; cc < 8; cc++) atomicAdd(&ag[cc], (vr[cc] + e[cc]) * w);
}

// conv1 beta gate + ReLU: wave per node, 8 ch/lane
__global__ void k_betarelu1(const float* __restrict__ agg, const float* __restrict__ xr,
                            const float* __restrict__ Wb, float* h, int Nn) {
  int t = threadIdx.x, wave = t >> 5, lane = t & 31;
  int n = blockIdx.x * 8 + wave;
  if (n >= Nn) return;
  int c0 = lane * 8;
  const float* ag = agg + (long)n * 256 + c0;
  const float* rp = xr + (long)n * 256 + c0;
  float s = 0.f;
#pragma unroll
  for (int cc = 0; cc < 8; cc++) {
    float o = ag[cc], r = rp[cc]; int c = c0 + cc;
    s += o * Wb[c] + r * Wb[256 + c] + (o - r) * Wb[512 + c];
  }
#pragma unroll
  for (int m = 1; m < 32; m <<= 1) s += __shfl_xor(s, m, 32);
  float beta = 1.f / (1.f + __expf(-s));
  float* hp = h + (long)n * 256 + c0;
#pragma unroll
  for (int cc = 0; cc < 8; cc++) {
    float y = beta * rp[cc] + (1.f - beta) * ag[cc];
    hp[cc] = y > 0.f ? y : 0.f;
  }
}

// ---------------------------------------------------------------------------
// fp32 WMMA GEMM: C[M,Nc] = A[M,K] * W[K,Nc] + bias. K<=256, K%4==0,
// M%16==0, Nc%16==0. Block = 8 waves; each wave owns one 16x16 output tile
// over full K (64x v_wmma_f32_16x16x4_f32). B column-tile staged in LDS.
// A frag (ISA 7.12.2): lane l -> row=l&15, kpair=2*(l>>4). B mirrored.
// D: lane l -> col=l&15, row r+8*(l>>4) in acc[r].
// ---------------------------------------------------------------------------
__global__ void k_gemm_wmma(const float* __restrict__ A, const float* __restrict__ W,
                            const float* __restrict__ bias, float* __restrict__ C,
                            int M, int K, int Nc) {
  __shared__ float Bs[256 * 16];
  int t = threadIdx.x;
  int col0 = blockIdx.y * 16;
  for (int r = t; r < K; r += 256) {
    const float* wr = W + (long)r * Nc + col0;
#pragma unroll
    for (int j = 0; j < 16; j++) Bs[r * 16 + j] = wr[j];
  }
  __syncthreads();
  int wave = t >> 5, lane = t & 31;
  int row0 = (blockIdx.x * 8 + wave) * 16;
  if (row0 + 16 > M) return;                  // wave-uniform: EXEC stays all-1s
  int lrow = lane & 15, lhi = lane >> 4;
  float bb = bias[col0 + lrow];
#if __has_builtin(__builtin_amdgcn_wmma_f32_16x16x4_f32)
  v8f acc = {};
  const float* arow = A + (long)(row0 + lrow) * K + 2 * lhi;
  for (int k = 0; k < K; k += 4) {
    float2 af = *(const float2*)(arow + k);   // A[row, k+2*lhi .. +1]
    v2f a; a.x = af.x; a.y = af.y;
    v2f b; b.x = Bs[(k + 2 * lhi) * 16 + lrow];
           b.y = Bs[(k + 2 * lhi + 1) * 16 + lrow];
    acc = __builtin_amdgcn_wmma_f32_16x16x4_f32(false, a, false, b,
                                                (short)0, acc, false, false);
  }
#pragma unroll
  for (int r = 0; r < 8; r++)
    C[(long)(row0 + r + 8 * lhi) * Nc + col0 + lrow] = acc[r] + bb;
#else
  float acc[8] = {0.f, 0.f, 0.f, 0.f, 0.f, 0.f, 0.f, 0.f};
  for (int k = 0; k < K; k++) {
    float bv = Bs[k * 16 + lrow];
#pragma unroll
    for (int r = 0; r < 8; r++)
      acc[r] += A[(long)(row0 + r + 8 * lhi) * K + k] * bv;
  }
#pragma unroll
  for (int r = 0; r < 8; r++)
    C[(long)(row0 + r + 8 * lhi) * Nc + col0 + lrow] = acc[r] + bb;
#endif
}

// ---------------------------------------------------------------------------
// conv2 edge passes (H=1, C=64): wave per edge, 2 ch/lane
// ---------------------------------------------------------------------------
__global__ void k_alpha2(const float* __restrict__ q2, const float* __restrict__ k2,
                         const float* __restrict__ ea, const int* __restrict__ srcA,
                         const int* __restrict__ dstA, const float* __restrict__ We,
                         float* alpha, int* amaxB, int Ee) {
  __shared__ float WeS[18 * 64];
  int t = threadIdx.x;
  for (int i = t; i < 18 * 64; i += 256) WeS[i] = We[i];
  __syncthreads();
  int wave = t >> 5, lane = t & 31;
  int eid = blockIdx.x * 8 + wave;
  if (eid >= Ee) return;
  int src = srcA[eid], dst = dstA[eid];
  float av = (lane < 18) ? ea[(long)eid * 18 + lane] : 0.f;
  int c0 = lane * 2;
  float e0 = 0.f, e1 = 0.f;
#pragma unroll
  for (int j = 0; j < 18; j++) {
    float a = __shfl(av, j, 32);
    e0 += a * WeS[j * 64 + c0];
    e1 += a * WeS[j * 64 + c0 + 1];
  }
  const float* kr = k2 + (long)src * 64 + c0;
  const float* qr = q2 + (long)dst * 64 + c0;
  float s = qr[0] * (kr[0] + e0) + qr[1] * (kr[1] + e1);
#pragma unroll
  for (int m = 1; m < 32; m <<= 1) s += __shfl_xor(s, m, 32);
  if (lane == 0) {
    float a = s * 0.125f;
    alpha[eid] = a;
    atomicMaxF((float*)&amaxB[dst], a);
  }
}

__global__ void k_expden2(float* alpha, const int* __restrict__ dstA,
                          const int* __restrict__ amaxB, float* den, int Ee) {
  long i = blockIdx.x * (long)blockDim.x + threadIdx.x;
  if (i >= Ee) return;
  int dst = dstA[i];
  float ex = __expf(alpha[i] - __int_as_float(amaxB[dst]));
  alpha[i] = ex;
  atomicAdd(&den[dst], ex);
}

__global__ void k_msg2(const float* __restrict__ v2, const float* __restrict__ ea,
                       const int* __restrict__ srcA, const int* __restrict__ dstA,
                       const float* __restrict__ We, const float* __restrict__ alpha,
                       const float* __restrict__ den, float* agg, int Ee) {
  __shared__ float WeS[18 * 64];
  int t = threadIdx.x;
  for (int i = t; i < 18 * 64; i += 256) WeS[i] = We[i];
  __syncthreads();
  int wave = t >> 5, lane = t & 31;
  int eid = blockIdx.x * 8 + wave;
  if (eid >= Ee) return;
  int src = srcA[eid], dst = dstA[eid];
  float av = (lane < 18) ? ea[(long)eid * 18 + lane] : 0.f;
  int c0 = lane * 2;
  float e0 = 0.f, e1 = 0.f;
#pragma unroll
  for (int j = 0; j < 18; j++) {
    float a = __shfl(av, j, 32);
    e0 += a * WeS[j * 64 + c0];
    e1 += a * WeS[j * 64 + c0 + 1];
  }
  float w = alpha[eid] / (den[dst] + 1e-16f);
  const float* vr = v2 + (long)src * 64 + c0;
  float* ag = agg + (long)dst * 64 + c0;
  atomicAdd(&ag[0], (vr[0] + e0) * w);
  atomicAdd(&ag[1], (vr[1] + e1) * w);
}

// conv2 beta gate + ReLU + fused global mean-pool accumulation
__global__ void k_betapool2(const float* __restrict__ agg, const float* __restrict__ xr,
                            const float* __restrict__ Wb, const int* __restrict__ batch,
                            float* psum, float* pcnt, int Nn) {
  int t = threadIdx.x, wave = t >> 5, lane = t & 31;
  int n = blockIdx.x * 8 + wave;
  if (n >= Nn) return;
  int c0 = lane * 2;
  const float* ag = agg + (long)n * 64 + c0;
  const float* rp = xr + (long)n * 64 + c0;
  float s = 0.f;
#pragma unroll
  for (int cc = 0; cc < 2; cc++) {
    float o = ag[cc], r = rp[cc]; int c = c0 + cc;
    s += o * Wb[c] + r * Wb[64 + c] + (o - r) * Wb[128 + c];
  }
#pragma unroll
  for (int m = 1; m < 32; m <<= 1) s += __shfl_xor(s, m, 32);
  float beta = 1.f / (1.f + __expf(-s));
  int g = batch[n];
#pragma unroll
  for (int cc = 0; cc < 2; cc++) {
    float y = beta * rp[cc] + (1.f - beta) * ag[cc];
    y = y > 0.f ? y : 0.f;
    atomicAdd(&psum[(long)g * 64 + c0 + cc], y);
  }
  if (lane == 0) atomicAdd(&pcnt[g], 1.f);
}

// final head: out[g, j] = (psum[g]/max(cnt,1)) @ Wlin + blin
__global__ void k_final(const float* __restrict__ psum, const float* __restrict__ pcnt,
                        const float* __restrict__ Wlin, const float* __restrict__ blin,
                        float* out, int Gg) {
  int t = blockIdx.x * blockDim.x + threadIdx.x;
  if (t >= Gg * 3) return;
  int g = t / 3, j = t % 3;
  float cnt = pcnt[g];
  cnt = cnt > 1.f ? cnt : 1.f;
  float inv = 1.f / cnt;
  float s = 0.f;
  for (int c = 0; c < 64; c++) s += psum[g * 64 + c] * inv * Wlin[c * 3 + j];
  out[t] = s + blin[j];
}

// ---------------------------------------------------------------------------
extern "C" void kernel_launch(void* const* d_in, const int* in_sizes, int n_in,
                              void* d_out, int out_size, void* d_ws, size_t ws_size,
                              hipStream_t stream) {
  const float* x    = (const float*)d_in[0];
  const int*   ei   = (const int*)d_in[1];
  const float* ea   = (const float*)d_in[2];
  const int*   bat  = (const int*)d_in[3];
  const float* Wq1  = (const float*)d_in[4];  const float* bq1 = (const float*)d_in[5];
  const float* Wk1  = (const float*)d_in[6];  const float* bk1 = (const float*)d_in[7];
  const float* Wv1  = (const float*)d_in[8];  const float* bv1 = (const float*)d_in[9];
  const float* We1  = (const float*)d_in[10];
  const float* Ws1  = (const float*)d_in[11]; const float* bs1 = (const float*)d_in[12];
  const float* Wb1  = (const float*)d_in[13];
  const float* Wq2  = (const float*)d_in[14]; const float* bq2 = (const float*)d_in[15];
  const float* Wk2  = (const float*)d_in[16]; const float* bk2 = (const float*)d_in[17];
  const float* Wv2  = (const float*)d_in[18]; const float* bv2 = (const float*)d_in[19];
  const float* We2  = (const float*)d_in[20];
  const float* Ws2  = (const float*)d_in[21]; const float* bs2 = (const float*)d_in[22];
  const float* Wb2  = (const float*)d_in[23];
  const float* Wlin = (const float*)d_in[24]; const float* blin = (const float*)d_in[25];
  float* out = (float*)d_out;

  const int Nn = in_sizes[0] / 2;      // x is [N,2]
  const int Ee = in_sizes[1] / 2;      // edge_index is [2,E]
  const int Gg = out_size / 3;
  const int* srcA = ei;
  const int* dstA = ei + Ee;

  // workspace arena (floats), ~217 MB total
  float* Wp = (float*)d_ws;
  size_t o = 0;
  float* q1     = Wp + o; o += (size_t)Nn * 256;
  float* k1     = Wp + o; o += (size_t)Nn * 256;
  float* v1     = Wp + o; o += (size_t)Nn * 256;
  float* xr1    = Wp + o; o += (size_t)Nn * 256;   // reused for q2|k2|v2|xr2
  float* agg1   = Wp + o; o += (size_t)Nn * 256;
  float* alpha1 = Wp + o; o += (size_t)Ee * 4;
  int*   amax1  = (int*)(Wp + o); o += (size_t)Nn * 4;
  float* den1   = Wp + o; o += (size_t)Nn * 4;
  float* agg2   = Wp + o; o += (size_t)Nn * 64;
  float* alpha2 = Wp + o; o += (size_t)Ee;
  int*   amax2  = (int*)(Wp + o); o += (size_t)Nn;
  float* den2   = Wp + o; o += (size_t)Nn;
  float* psum   = Wp + o; o += (size_t)Gg * 64;
  float* pcnt   = Wp + o; o += (size_t)Gg;
  (void)ws_size; (void)n_in;

  float* h1  = q1;                      // safe reuse: q1 dead after k_alpha1
  float* q2  = xr1;                     // safe reuse: xr1 dead after k_betarelu1
  float* k2  = xr1 + (size_t)Nn * 64;
  float* v2  = xr1 + (size_t)2 * Nn * 64;
  float* xr2 = xr1 + (size_t)3 * Nn * 64;

  dim3 b256(256);
  unsigned eb = (unsigned)((Ee + 7) / 8);
  unsigned nb = (unsigned)((Nn + 7) / 8);

  k_init<<<2048, b256, 0, stream>>>(agg1, den1, amax1, agg2, den2, amax2,
                                    psum, pcnt, Nn, Gg);
  k_nodeproj1<<<(unsigned)(((long)Nn * 256 + 255) / 256), b256, 0, stream>>>(
      x, Wq1, bq1, Wk1, bk1, Wv1, bv1, Ws1, bs1, q1, k1, v1, xr1, Nn);
  k_alpha1<<<eb, b256, 0, stream>>>(q1, k1, ea, srcA, dstA, We1, alpha1, amax1, Ee);
  k_expden1<<<(unsigned)(((long)Ee * 4 + 255) / 256), b256, 0, stream>>>(
      alpha1, dstA, amax1, den1, Ee);
  k_msg1<<<eb, b256, 0, stream>>>(v1, ea, srcA, dstA, We1, alpha1, den1, agg1, Ee);
  k_betarelu1<<<nb, b256, 0, stream>>>(agg1, xr1, Wb1, h1, Nn);

  {
    dim3 g((unsigned)((Nn / 16 + 7) / 8), 4);   // 2500 row tiles, 4 col tiles
    k_gemm_wmma<<<g, b256, 0, stream>>>(h1, Wq2, bq2, q2, Nn, 256, 64);
    k_gemm_wmma<<<g, b256, 0, stream>>>(h1, Wk2, bk2, k2, Nn, 256, 64);
    k_gemm_wmma<<<g, b256, 0, stream>>>(h1, Wv2, bv2, v2, Nn, 256, 64);
    k_gemm_wmma<<<g, b256, 0, stream>>>(h1, Ws2, bs2, xr2, Nn, 256, 64);
  }

  k_alpha2<<<eb, b256, 0, stream>>>(q2, k2, ea, srcA, dstA, We2, alpha2, amax2, Ee);
  k_expden2<<<(unsigned)((Ee + 255) / 256), b256, 0, stream>>>(alpha2, dstA, amax2, den2, Ee);
  k_msg2<<<eb, b256, 0, stream>>>(v2, ea, srcA, dstA, We2, alpha2, den2, agg2, Ee);
  k_betapool2<<<nb, b256, 0, stream>>>(agg2, xr2, Wb2, bat, psum, pcnt, Nn);
  k_final<<<1, b256, 0, stream>>>(psum, pcnt, Wlin, blin, out, Gg);
}